// IndexSelect_8761733284512
// MI455X (gfx1250) — compile-verified
//
#include <hip/hip_runtime.h>
#include <math.h>

#define B_    16
#define NH    64
#define NPTS  2048
#define NEIGHS 20
#define K_SEL 512
#define EPSV  1e-5f

typedef float v2f __attribute__((ext_vector_type(2)));
typedef float v8f __attribute__((ext_vector_type(8)));

// D = A(16x4 f32) * B(4x16 f32) + C(16x16 f32), full fp32 WMMA.
__device__ __forceinline__ v8f wmma4(v2f a, v2f b, v8f c) {
  return __builtin_amdgcn_wmma_f32_16x16x4_f32(false, a, false, b, (short)0, c,
                                               false, false);
}

__device__ __forceinline__ float sigm(float x) { return 1.0f / (1.0f + expf(-x)); }

// JAX top_k ordering: larger value first; ties -> lower index first.
__device__ __forceinline__ bool better(float v1, int i1, float v2, int i2) {
  return (v1 > v2) || (v1 == v2 && i1 < i2);
}

// ---------------------------------------------------------------------------
// K1: h = relu(BN(W_fc @ seq + b)).  grid(z)=0: seq1 -> h1 ; =1: seq1[:,perm] -> h2
// One 128-thread block = 4 waves; wave w owns output-channel tile [16w,16w+16)
// for the same 16 points.  K=64 swept in 16 WMMA 16x16x4 steps.
// ---------------------------------------------------------------------------
__global__ __launch_bounds__(128) void k_fc(
    const float* __restrict__ seq1, const int* __restrict__ perm,
    const float* __restrict__ Wfc, const float* __restrict__ bfc,
    const float* __restrict__ g1, const float* __restrict__ be1,
    const float* __restrict__ mu1, const float* __restrict__ va1,
    float* __restrict__ h1, float* __restrict__ h2) {
  const int tid  = threadIdx.x;
  const int wave = tid >> 5, lane = tid & 31;
  const int row  = lane & 15, hi = lane >> 4;
  const int o0 = wave * 16;
  const int n0 = blockIdx.x * 16;
  const int b  = blockIdx.y;
  const int which = blockIdx.z;
  const float* s = seq1 + (size_t)b * NH * NPTS;
  int p = n0 + row;
  if (which) p = perm[p];
  v8f acc = {};
#pragma unroll
  for (int k0 = 0; k0 < NH; k0 += 4) {
    v2f a, bb;
    a.x  = Wfc[(o0 + row) * NH + k0 + 2 * hi];      // A: lane=M, half/vgpr=K
    a.y  = Wfc[(o0 + row) * NH + k0 + 2 * hi + 1];
    bb.x = s[(size_t)(k0 + 2 * hi) * NPTS + p];      // B: lane=N, half/vgpr=K
    bb.y = s[(size_t)(k0 + 2 * hi + 1) * NPTS + p];
    acc = wmma4(a, bb, acc);
  }
  float* h = (which ? h2 : h1) + (size_t)b * NH * NPTS;
  const int n = n0 + row;                            // C: lane=N
#pragma unroll
  for (int j = 0; j < 8; ++j) {
    int ch = o0 + j + 8 * hi;                        // C: vgpr+half=M
    float sc = g1[ch] * rsqrtf(va1[ch] + EPSV);
    float v  = (acc[j] + bfc[ch] - mu1[ch]) * sc + be1[ch];
    h[(size_t)ch * NPTS + n] = fmaxf(v, 0.0f);
  }
}

// ---------------------------------------------------------------------------
// K2: x2[b][n] = sum_c h1[b][c][n]^2
// ---------------------------------------------------------------------------
__global__ __launch_bounds__(256) void k_x2(const float* __restrict__ h1,
                                            float* __restrict__ x2) {
  int i = blockIdx.x * 256 + threadIdx.x;   // 0..B*NPTS-1
  int b = i / NPTS, n = i % NPTS;
  const float* h = h1 + (size_t)b * NH * NPTS + n;
  float s = 0.f;
#pragma unroll
  for (int c = 0; c < NH; ++c) { float v = h[(size_t)c * NPTS]; s += v * v; }
  x2[i] = s;
}

// ---------------------------------------------------------------------------
// K3: fused Gram-matrix WMMA + top-20 KNN.  One WG = 32 query rows; the 8
// waves are tiled 2 (row-tiles) x 4 (col-tiles) so every streamed B column
// tile is consumed by two row tiles -> L2 read traffic halved vs a 16-row WG
// (~0.5 GB total instead of ~1 GB).  Per sweep step the WG produces a 32x64
// distance tile in LDS; 256 threads (8 per row) keep register-resident sorted
// top-20 lists (fully unrolled bubble-insert -> stays in VGPRs); 8-way sorted
// merge per row at the end.  The 16x2048x2048 distance matrix never touches
// HBM (saves ~536 MB of traffic ~ 23 us at 23.3 TB/s).
// ---------------------------------------------------------------------------
__global__ __launch_bounds__(256) void k_knn(
    const float* __restrict__ h1, const float* __restrict__ x2,
    int* __restrict__ nidx) {
  __shared__ float dtile[32][65];          // padded: no bank conflicts on scan
  __shared__ float mv[32][8][NEIGHS];
  __shared__ int   mi[32][8][NEIGHS];
  const int tid  = threadIdx.x;
  const int wave = tid >> 5, lane = tid & 31;
  const int row  = lane & 15, hi = lane >> 4;
  const int wr = wave >> 2;                // row-tile 0/1
  const int wc = wave & 3;                 // col-tile 0..3
  const int n0 = blockIdx.x * 32;
  const int b  = blockIdx.y;
  const float* h  = h1 + (size_t)b * NH * NPTS;
  const float* xs = x2 + b * NPTS;

  // Cache A fragments (this wave's 16 rows, all 64 channels): 16 x v2f VGPRs.
  const int nrow = n0 + wr * 16 + row;
  v2f areg[16];
#pragma unroll
  for (int s = 0; s < 16; ++s) {
    int k0 = s * 4;
    areg[s].x = h[(size_t)(k0 + 2 * hi) * NPTS + nrow];
    areg[s].y = h[(size_t)(k0 + 2 * hi + 1) * NPTS + nrow];
  }
  float x2n[8];
#pragma unroll
  for (int j = 0; j < 8; ++j) x2n[j] = xs[n0 + wr * 16 + j + 8 * hi];

  float tv[NEIGHS]; int ti[NEIGHS];
#pragma unroll
  for (int q = 0; q < NEIGHS; ++q) { tv[q] = -3.4e38f; ti[q] = 0x7fffffff; }

  const int selRow = tid >> 3, selSlot = tid & 7;   // 32 rows x 8 slots

  for (int mb = 0; mb < NPTS; mb += 64) {
    const int m0 = mb + wc * 16;
    v8f acc = {};
#pragma unroll
    for (int s = 0; s < 16; ++s) {
      int k0 = s * 4;
      v2f bb;
      bb.x = h[(size_t)(k0 + 2 * hi) * NPTS + m0 + row];
      bb.y = h[(size_t)(k0 + 2 * hi + 1) * NPTS + m0 + row];
      acc = wmma4(areg[s], bb, acc);
    }
    float x2m = xs[m0 + row];
#pragma unroll
    for (int j = 0; j < 8; ++j)
      dtile[wr * 16 + j + 8 * hi][wc * 16 + row] =
          2.0f * acc[j] - x2n[j] - x2m;
    __syncthreads();

    // Thread (row, slot) scans 8 columns of its row in this 32x64 tile.
#pragma unroll
    for (int cc = 0; cc < 8; ++cc) {
      int col = selSlot * 8 + cc;
      float v = dtile[selRow][col];
      int   m = mb + col;
      if (better(v, m, tv[NEIGHS - 1], ti[NEIGHS - 1])) {
        float cv = v; int ci = m;
#pragma unroll
        for (int q = 0; q < NEIGHS; ++q) {
          bool bt = better(cv, ci, tv[q], ti[q]);
          float ov = tv[q]; int oi = ti[q];
          if (bt) { tv[q] = cv; ti[q] = ci; cv = ov; ci = oi; }
        }
      }
    }
    __syncthreads();
  }

#pragma unroll
  for (int q = 0; q < NEIGHS; ++q) {
    mv[selRow][selSlot][q] = tv[q];
    mi[selRow][selSlot][q] = ti[q];
  }
  __syncthreads();
  if (tid < 32) {  // 8-way merge of sorted lists, one thread per query row
    int pos[8];
#pragma unroll
    for (int s = 0; s < 8; ++s) pos[s] = 0;
    int* out = nidx + ((size_t)(b * NPTS + n0 + tid)) * NEIGHS;
    for (int k = 0; k < NEIGHS; ++k) {
      float bvv = -3.4e38f; int bii = 0x7fffffff; int bs = 0;
#pragma unroll
      for (int s = 0; s < 8; ++s) {
        float v = mv[tid][s][pos[s]]; int i = mi[tid][s][pos[s]];
        if (better(v, i, bvv, bii)) { bvv = v; bii = i; bs = s; }
      }
      pos[bs]++;
      out[k] = bii;
    }
  }
}

// ---------------------------------------------------------------------------
// K4: edge conv.  One wave per point: feat(32x128, rows 20..31 don't matter,
// masked out of the max) x W_ec^T(128x64) via WMMA, fused BN + leaky-ReLU +
// max over neighbors + sigmoid -> X[b][n][64].
// ---------------------------------------------------------------------------
__global__ __launch_bounds__(64) void k_edge(
    const float* __restrict__ h1, const int* __restrict__ nidx,
    const float* __restrict__ Wec,
    const float* __restrict__ g2, const float* __restrict__ be2,
    const float* __restrict__ mu2, const float* __restrict__ va2,
    float* __restrict__ X) {
  __shared__ float feat[2][32][128];
  const int tid  = threadIdx.x;
  const int wave = tid >> 5, lane = tid & 31;
  const int row  = lane & 15, hi = lane >> 4;
  const int n = blockIdx.x * 2 + wave;
  const int b = blockIdx.y;
  const float* h  = h1 + (size_t)b * NH * NPTS;
  const int*   ni = nidx + ((size_t)(b * NPTS + n)) * NEIGHS;
  const int c0 = lane, c1 = lane + 32;
  float ctr0 = h[(size_t)c0 * NPTS + n];
  float ctr1 = h[(size_t)c1 * NPTS + n];
  for (int k = 0; k < NEIGHS; ++k) {
    int nb = ni[k];
    feat[wave][k][c0]      = h[(size_t)c0 * NPTS + nb] - ctr0;
    feat[wave][k][64 + c0] = ctr0;
    feat[wave][k][c1]      = h[(size_t)c1 * NPTS + nb] - ctr1;
    feat[wave][k][64 + c1] = ctr1;
  }
  __syncthreads();
  float* Xp = X + ((size_t)(b * NPTS + n)) * NH;
#pragma unroll
  for (int t = 0; t < 4; ++t) {
    const int o = t * 16 + row;     // output channel for this lane (N dim)
    v8f acc0 = {}, acc1 = {};
#pragma unroll 4
    for (int s = 0; s < 32; ++s) {
      int k0 = s * 4;
      v2f a0, a1, bb;
      a0.x = feat[wave][row][k0 + 2 * hi];
      a0.y = feat[wave][row][k0 + 2 * hi + 1];
      a1.x = feat[wave][16 + row][k0 + 2 * hi];
      a1.y = feat[wave][16 + row][k0 + 2 * hi + 1];
      bb.x = Wec[o * 128 + k0 + 2 * hi];
      bb.y = Wec[o * 128 + k0 + 2 * hi + 1];
      acc0 = wmma4(a0, bb, acc0);   // neighbor rows 0..15
      acc1 = wmma4(a1, bb, acc1);   // neighbor rows 16..31 (only 16..19 valid)
    }
    float s2 = g2[o] * rsqrtf(va2[o] + EPSV);
    float m2 = mu2[o], bt2 = be2[o];
    float pm = -3.4e38f;
#pragma unroll
    for (int j = 0; j < 8; ++j) {
      float y = (acc0[j] - m2) * s2 + bt2;
      y = (y >= 0.f) ? y : 0.2f * y;
      pm = fmaxf(pm, y);
    }
    if (hi == 0) {                  // acc1 rows 16+j (j<4) are the valid ones
#pragma unroll
      for (int j = 0; j < 4; ++j) {
        float y = (acc1[j] - m2) * s2 + bt2;
        y = (y >= 0.f) ? y : 0.2f * y;
        pm = fmaxf(pm, y);
      }
    }
    pm = fmaxf(pm, __shfl_xor(pm, 16, 32));   // combine lane halves
    if (hi == 0) Xp[o] = sigm(pm);
  }
}

// ---------------------------------------------------------------------------
// K5: sc = h^T W_b X + b_b for h1 (->ret[:,0:N], scores) and h2 (->ret[:,N:2N]).
// One wave per point; lane owns columns d and d+32 of W_b.
// ---------------------------------------------------------------------------
__global__ __launch_bounds__(256) void k_score(
    const float* __restrict__ h1, const float* __restrict__ h2,
    const float* __restrict__ Wb, const float* __restrict__ bb_,
    const float* __restrict__ X,
    float* __restrict__ ret, float* __restrict__ scores) {
  const int tid  = threadIdx.x;
  const int wave = tid >> 5, lane = tid & 31;
  const int n = blockIdx.x * 8 + wave;
  const int b = blockIdx.y;
  const float* ha = h1 + (size_t)b * NH * NPTS + n;
  const float* hb = h2 + (size_t)b * NH * NPTS + n;
  const float* Xp = X + ((size_t)(b * NPTS + n)) * NH;
  const int d0 = lane, d1 = lane + 32;
  float t1a = 0.f, t1b = 0.f, t2a = 0.f, t2b = 0.f;
#pragma unroll 8
  for (int c = 0; c < NH; ++c) {
    float w0 = Wb[c * NH + d0], w1 = Wb[c * NH + d1];
    float va = ha[(size_t)c * NPTS], vb = hb[(size_t)c * NPTS];
    t1a += va * w0; t1b += va * w1;
    t2a += vb * w0; t2b += vb * w1;
  }
  float part1 = t1a * Xp[d0] + t1b * Xp[d1];
  float part2 = t2a * Xp[d0] + t2b * Xp[d1];
#pragma unroll
  for (int off = 16; off > 0; off >>= 1) {
    part1 += __shfl_xor(part1, off, 32);
    part2 += __shfl_xor(part2, off, 32);
  }
  if (lane == 0) {
    float bbv = bb_[0];
    float s1 = part1 + bbv, s2 = part2 + bbv;
    ret[(size_t)b * (2 * NPTS) + n]        = s1;
    ret[(size_t)b * (2 * NPTS) + NPTS + n] = s2;
    scores[b * NPTS + n] = sigm(s1);
  }
}

// ---------------------------------------------------------------------------
// K6: per-batch bitonic sort of 2048 (score, idx) pairs in LDS (descending,
// lower-index tiebreak == jax top_k), then emit top-512 values/idx and do the
// seq1/xyz gathers + scaling straight into d_out.
// ---------------------------------------------------------------------------
__global__ __launch_bounds__(1024) void k_topk(
    const float* __restrict__ scores, const float* __restrict__ seq1,
    const float* __restrict__ xyz,
    float* __restrict__ seq_out, float* __restrict__ val_out,
    float* __restrict__ idx_out, float* __restrict__ xyzs_out,
    float* __restrict__ xyzo_out) {
  __shared__ float key[NPTS];
  __shared__ int   kid[NPTS];
  const int tid = threadIdx.x;
  const int b   = blockIdx.x;
  for (int t = tid; t < NPTS; t += 1024) { key[t] = scores[b * NPTS + t]; kid[t] = t; }
  for (unsigned k = 2; k <= NPTS; k <<= 1) {
    for (unsigned j = k >> 1; j > 0; j >>= 1) {
      __syncthreads();
      for (unsigned t = tid; t < NPTS; t += 1024) {
        unsigned x = t ^ j;
        if (x > t) {
          float v1 = key[t], v2 = key[x];
          int   i1 = kid[t], i2 = kid[x];
          bool desc = ((t & k) == 0);
          bool sw = desc ? better(v2, i2, v1, i1) : better(v1, i1, v2, i2);
          if (sw) { key[t] = v2; kid[t] = i2; key[x] = v1; kid[x] = i1; }
        }
      }
    }
  }
  __syncthreads();
  if (tid < K_SEL) {
    const int j  = tid;
    float val = key[j];
    int   id  = kid[j];
    val_out[b * K_SEL + j] = val;
    idx_out[b * K_SEL + j] = (float)id;
    const float* s  = seq1 + (size_t)b * NH * NPTS;
    float* so = seq_out + (size_t)b * NH * K_SEL;
#pragma unroll 8
    for (int c = 0; c < NH; ++c)
      so[c * K_SEL + j] = s[(size_t)c * NPTS + id] * val;
    const float* xz = xyz + (size_t)b * 3 * NPTS;
#pragma unroll
    for (int c = 0; c < 3; ++c) {
      float v = xz[(size_t)c * NPTS + id];
      xyzs_out[((size_t)b * 3 + c) * K_SEL + j] = v;
      xyzo_out[((size_t)b * 3 + c) * K_SEL + j] = v * val;
    }
  }
}

extern "C" void kernel_launch(void* const* d_in, const int* in_sizes, int n_in,
                              void* d_out, int out_size, void* d_ws, size_t ws_size,
                              hipStream_t stream) {
  (void)in_sizes; (void)n_in; (void)out_size; (void)ws_size;
  const float* xyz  = (const float*)d_in[0];
  const float* seq1 = (const float*)d_in[1];
  const int*   perm = (const int*)d_in[2];
  const float* Wfc  = (const float*)d_in[3];
  const float* bfc  = (const float*)d_in[4];
  const float* g1   = (const float*)d_in[5];
  const float* be1  = (const float*)d_in[6];
  const float* mu1  = (const float*)d_in[7];
  const float* va1  = (const float*)d_in[8];
  const float* Wec  = (const float*)d_in[9];
  const float* g2   = (const float*)d_in[10];
  const float* be2  = (const float*)d_in[11];
  const float* mu2  = (const float*)d_in[12];
  const float* va2  = (const float*)d_in[13];
  const float* Wb   = (const float*)d_in[14];
  const float* bb   = (const float*)d_in[15];

  char* ws = (char*)d_ws;
  size_t off = 0;
  float* h1 = (float*)(ws + off); off += (size_t)B_ * NH * NPTS * 4;
  float* h2 = (float*)(ws + off); off += (size_t)B_ * NH * NPTS * 4;
  float* x2 = (float*)(ws + off); off += (size_t)B_ * NPTS * 4;
  float* X  = (float*)(ws + off); off += (size_t)B_ * NPTS * NH * 4;
  int* nidx = (int*)(ws + off);   off += (size_t)B_ * NPTS * NEIGHS * 4;
  float* scores = (float*)(ws + off); off += (size_t)B_ * NPTS * 4;

  float* out = (float*)d_out;
  float* seq_out  = out;                                    // B*NH*K_SEL
  float* val_out  = seq_out + (size_t)B_ * NH * K_SEL;      // B*K_SEL
  float* idx_out  = val_out + (size_t)B_ * K_SEL;           // B*K_SEL
  float* ret_out  = idx_out + (size_t)B_ * K_SEL;           // B*2*NPTS
  float* xyzs_out = ret_out + (size_t)B_ * 2 * NPTS;        // B*3*K_SEL
  float* xyzo_out = xyzs_out + (size_t)B_ * 3 * K_SEL;      // B*3*K_SEL

  k_fc<<<dim3(NPTS / 16, B_, 2), 128, 0, stream>>>(seq1, perm, Wfc, bfc, g1, be1,
                                                   mu1, va1, h1, h2);
  k_x2<<<dim3(B_ * NPTS / 256), 256, 0, stream>>>(h1, x2);
  k_knn<<<dim3(NPTS / 32, B_), 256, 0, stream>>>(h1, x2, nidx);
  k_edge<<<dim3(NPTS / 2, B_), 64, 0, stream>>>(h1, nidx, Wec, g2, be2, mu2, va2, X);
  k_score<<<dim3(NPTS / 8, B_), 256, 0, stream>>>(h1, h2, Wb, bb, X, ret_out, scores);
  k_topk<<<dim3(B_), 1024, 0, stream>>>(scores, seq1, xyz, seq_out, val_out,
                                        idx_out, xyzs_out, xyzo_out);
}